// LinearModel_37203006717978
// MI455X (gfx1250) — compile-verified
//
#include <hip/hip_runtime.h>

typedef __attribute__((ext_vector_type(2))) float v2f;
typedef __attribute__((ext_vector_type(4))) float v4f;
typedef __attribute__((ext_vector_type(8))) float v8f;

#define KDIM            128
#define ROWS_PER_WAVE   16
#define WAVES_PER_BLOCK 8
#define LDS_STRIDE      132   // 128 + 4 pad: keeps b128 stores 16B-aligned and ds_load_b64 bank-conflict-free

// out[row] = dot(A[row, 0:128], theta[0:128]) for 1M rows.
// One wave produces 16 outputs per tile via 32 chained V_WMMA_F32_16X16X4_F32:
//   A-frag  = data tile (16 x 4 per step),  B-frag = theta broadcast across N.
// Every column of D is then the 16 dot products.
__global__ __launch_bounds__(256) void linmodel_matvec_wmma(
    const float* __restrict__ A, const float* __restrict__ theta,
    float* __restrict__ out, int numTiles) {
  __shared__ float smemA[WAVES_PER_BLOCK * ROWS_PER_WAVE * LDS_STRIDE];
  __shared__ float smemT[KDIM];

  const int tid  = threadIdx.x;
  const int lane = tid & 31;
  const int wave = tid >> 5;

  // Stage theta once per block (broadcast reads later, no conflicts).
  if (tid < KDIM) smemT[tid] = theta[tid];

  const int tile   = blockIdx.x * WAVES_PER_BLOCK + wave;  // wave-uniform
  const bool active = tile < numTiles;                     // wave-uniform -> EXEC stays full

  float* myLds = smemA + wave * ROWS_PER_WAVE * LDS_STRIDE;

  if (active) {
    // Cooperative, fully coalesced staging: each global_load_b128 covers one
    // entire 512B row (32 lanes x float4).
    const size_t rowBase = (size_t)tile * ROWS_PER_WAVE;
#pragma unroll
    for (int r = 0; r < ROWS_PER_WAVE; ++r) {
      const float* src = A + (rowBase + r) * KDIM + lane * 4;
      v4f v = *(const v4f*)src;
      *(v4f*)(myLds + r * LDS_STRIDE + lane * 4) = v;
    }
  }
  __syncthreads();  // orders LDS stores vs. cross-lane LDS reads (all threads reach this)

  if (active) {
    const int m  = lane & 15;   // which of the 16 rows this lane supplies to A-frag
    const int hi = lane >> 4;   // lane half selects K pair {0,1} vs {2,3} within each chunk
    const float* rowp = myLds + m * LDS_STRIDE;

    v8f c = {};
#pragma unroll
    for (int kc = 0; kc < KDIM; kc += 4) {
      const int kk = kc + 2 * hi;
      // A-frag (32-bit 16x4): VGPR0 = A[m, kc+2*hi], VGPR1 = A[m, kc+2*hi+1]  -> one ds_load_b64
      v2f a = *(const v2f*)(rowp + kk);
      // B-frag (4x16, theta broadcast over N): same K->(vgpr,half) striping as A/C layouts
      v2f b = *(const v2f*)(smemT + kk);
      c = __builtin_amdgcn_wmma_f32_16x16x4_f32(
          /*neg_a=*/false, a, /*neg_b=*/false, b,
          /*c_mod=*/(short)0, c, /*reuse_a=*/false, /*reuse_b=*/false);
    }

    // D[m,n] identical for all n. Lane 0 holds rows 0..7 in c[0..7] (M=j, lanes 0-15);
    // lane 16 holds rows 8..15 (M=j+8 lives in lanes 16-31). Two b128 stores.
    if (m == 0) {
      const size_t o = (size_t)tile * ROWS_PER_WAVE + hi * 8;
      v4f lo = {c[0], c[1], c[2], c[3]};
      v4f hiq = {c[4], c[5], c[6], c[7]};
      *(v4f*)(out + o)     = lo;
      *(v4f*)(out + o + 4) = hiq;
    }
  }
}

extern "C" void kernel_launch(void* const* d_in, const int* in_sizes, int n_in,
                              void* d_out, int out_size, void* d_ws, size_t ws_size,
                              hipStream_t stream) {
  const float* A     = (const float*)d_in[0];   // link_features (N*M*128)
  const float* theta = (const float*)d_in[2];   // theta (128)
  float* out         = (float*)d_out;           // (N*M)

  const int totalRows = out_size;               // 1024*1024
  const int numTiles  = totalRows / ROWS_PER_WAVE;  // 16 rows per wave-tile
  const int blocks    = (numTiles + WAVES_PER_BLOCK - 1) / WAVES_PER_BLOCK;

  linmodel_matvec_wmma<<<blocks, 256, 0, stream>>>(A, theta, out, numTiles);
}